// CompgcnLP_11209864642821
// MI455X (gfx1250) — compile-verified
//
#include <hip/hip_runtime.h>
#include <math.h>

typedef float v2f __attribute__((ext_vector_type(2)));
typedef float v8f __attribute__((ext_vector_type(8)));

#define NENT   100000
#define DHID   128
#define NEDGE  800000
#define NTRI   4096
#define NRELV  401
#define NBASE  50
#define MTILES (NENT / 16)          // 6250, exact
#define BN_EPS 1e-5f

// ---- workspace layout (in floats) ----
static constexpr size_t SZ_A      = (size_t)3 * NENT * DHID;   // 38,400,000
static constexpr size_t SZ_STATS  = 256;                       // sum[128], sumsq[128]
static constexpr size_t OFF_A     = 0;
static constexpr size_t OFF_STATS = OFF_A + SZ_A;
static constexpr size_t OFF_AGG   = OFF_STATS + SZ_STATS;
static constexpr size_t OFF_X1    = OFF_AGG + (size_t)NENT * DHID;
static constexpr size_t OFF_X2    = OFF_X1  + (size_t)NENT * DHID;
static constexpr size_t OFF_R0    = OFF_X2  + (size_t)NENT * DHID;
static constexpr size_t OFF_R1    = OFF_R0  + (size_t)NRELV * DHID;
static constexpr size_t OFF_R2    = OFF_R1  + (size_t)NRELV * DHID;
// total ~77M floats (~308 MB) of d_ws

#define WT_STRIDE 132   // padded LDS stride (floats) -> conflict-free b64 B reads
#define GEMM_LDS_BYTES ((128 * WT_STRIDE + 4 * 16 * 128) * 4)  // 100,352 B

// ---------------------------------------------------------------------------
// r0 = concat(coeff@bases, -(coeff@bases), self_rel)
__global__ void rel_init_kernel(const float* __restrict__ coeff,
                                const float* __restrict__ bases,
                                const float* __restrict__ selfr,
                                float* __restrict__ R) {
    int i = blockIdx.x * blockDim.x + threadIdx.x;
    if (i >= NRELV * DHID) return;
    int rel = i >> 7, c = i & 127;
    if (rel == 400) { R[i] = selfr[c]; return; }
    int rr = (rel >= 200) ? rel - 200 : rel;
    float s = 0.f;
    #pragma unroll 5
    for (int b = 0; b < NBASE; ++b) s += coeff[rr * NBASE + b] * bases[b * DHID + c];
    R[i] = (rel >= 200) ? -s : s;
}

// Rout = R @ Wr   (401x128 @ 128x128 — tiny)
__global__ void rel_mm_kernel(const float* __restrict__ R,
                              const float* __restrict__ Wr,
                              float* __restrict__ Rout) {
    int i = blockIdx.x * blockDim.x + threadIdx.x;
    if (i >= NRELV * DHID) return;
    int row = i >> 7, c = i & 127;
    float s = 0.f;
    for (int k = 0; k < DHID; ++k) s += R[row * DHID + k] * Wr[k * DHID + c];
    Rout[i] = s;
}

// grid-stride float4 zero fill
__global__ void zero_kernel(float4* __restrict__ p, size_t n4) {
    size_t i = (size_t)blockIdx.x * blockDim.x + threadIdx.x;
    size_t stride = (size_t)gridDim.x * blockDim.x;
    float4 z = {0.f, 0.f, 0.f, 0.f};
    for (; i < n4; i += stride) p[i] = z;
}

// A[y[e]][dst[e]][:] += x[src[e]][:] - r[edge_type[e]][:]   (one wave per edge)
// Lane L covers cols 4L..4L+3: two b128 loads + 4 consecutive-address atomics.
// The wave's atomic burst covers one contiguous 512B window per edge row.
__global__ void scatter_kernel(const float* __restrict__ X,
                               const float* __restrict__ R,
                               const int* __restrict__ eidx,   // [2][NEDGE]
                               const int* __restrict__ etype,  // [NEDGE]
                               const int* __restrict__ ey,     // [NEDGE]
                               float* __restrict__ Abuf) {
    int e = blockIdx.x * (blockDim.x >> 5) + (threadIdx.x >> 5);
    if (e >= NEDGE) return;
    int lane = threadIdx.x & 31;
    int src = eidx[e];
    int dst = eidx[NEDGE + e];
    int et  = etype[e];
    int k   = ey[e];
    const int c = lane << 2;
    const float4 xv = *(const float4*)(X + (size_t)src * DHID + c);
    const float4 rv = *(const float4*)(R + (size_t)et  * DHID + c);
    float* outp = Abuf + ((size_t)k * NENT + (size_t)dst) * DHID + c;
    atomicAdd(outp + 0, xv.x - rv.x);
    atomicAdd(outp + 1, xv.y - rv.y);
    atomicAdd(outp + 2, xv.z - rv.z);
    atomicAdd(outp + 3, xv.w - rv.w);
}

// AGG = A_cat(100000 x 384) @ W_cat(384 x 128) via V_WMMA_F32_16X16X4_F32.
// Each wave owns one 16-row M tile x full N=128 (8 x v8f accumulators).
// W slab (128x128) staged transposed+padded in LDS; D staged in LDS for
// coalesced float4 stores.
__global__ void __launch_bounds__(128)
gemm_wmma_kernel(const float* __restrict__ Abuf,   // [3][NENT][128]
                 const float* __restrict__ W,      // [3][128][128]
                 float* __restrict__ AGG) {        // [NENT][128]
    extern __shared__ float smem[];
    float* WT  = smem;                         // 128 * WT_STRIDE
    float* STG = smem + 128 * WT_STRIDE;       // 4 waves * 16*128

    const int tid   = threadIdx.x;
    const int lane  = tid & 31;
    const int wave  = tid >> 5;
    const int mtile = blockIdx.x * 4 + wave;
    const int mrow  = lane & 15;               // M (for A) / N-within-tile (for B)
    const int koff  = (lane >> 4) << 1;        // 0 or 2 : K sub-offset per lane half
    const bool active = (mtile < MTILES);      // wave-uniform -> EXEC stays all-1s

    v8f acc[8] = {};

    for (int kk = 0; kk < 3; ++kk) {
        __syncthreads();
        // stage W[kk] transposed: WT[j][d] = W[kk][d][j]
        for (int i = tid; i < 128 * 32; i += 128) {
            const int d  = i >> 5;
            const int j4 = (i & 31) << 2;
            const float4 w = *(const float4*)(W + ((size_t)(kk * 128 + d)) * 128 + j4);
            WT[(j4 + 0) * WT_STRIDE + d] = w.x;
            WT[(j4 + 1) * WT_STRIDE + d] = w.y;
            WT[(j4 + 2) * WT_STRIDE + d] = w.z;
            WT[(j4 + 3) * WT_STRIDE + d] = w.w;
        }
        __syncthreads();

        if (active) {
            const float* Arow =
                Abuf + ((size_t)kk * NENT + (size_t)mtile * 16 + mrow) * DHID;
            for (int s = 0; s < 32; ++s) {     // 32 K-steps of 4 per slab
                const v2f a = *(const v2f*)(Arow + s * 4 + koff);
                #pragma unroll
                for (int n = 0; n < 8; ++n) {
                    const v2f b =
                        *(const v2f*)(WT + (n * 16 + mrow) * WT_STRIDE + s * 4 + koff);
                    acc[n] = __builtin_amdgcn_wmma_f32_16x16x4_f32(
                        false, a, false, b, (short)0, acc[n], false, false);
                }
            }
        }
    }

    if (active) {
        // D layout: acc[n][v] holds M = v + (lane>=16 ? 8:0), N = n*16 + (lane&15)
        float* stg = STG + wave * (16 * 128);
        const int rbase = (lane >> 4) << 3;
        #pragma unroll
        for (int n = 0; n < 8; ++n)
            #pragma unroll
            for (int v = 0; v < 8; ++v)
                stg[(rbase + v) * 128 + n * 16 + mrow] = acc[n][v];
        asm volatile("s_wait_dscnt 0" ::: "memory");
        float* Orow = AGG + (size_t)mtile * 16 * DHID;
        #pragma unroll
        for (int r = 0; r < 16; ++r) {
            const float4 val = *(const float4*)(stg + r * 128 + lane * 4);
            *(float4*)(Orow + (size_t)r * 128 + lane * 4) = val;             // coalesced
        }
    }
}

// column sums / sums of squares for batchnorm
__global__ void bn_stats_kernel(const float* __restrict__ AGG,
                                float* __restrict__ stats) {
    int c = threadIdx.x & 127;
    int row = blockIdx.x * (blockDim.x >> 7) + (threadIdx.x >> 7);
    int stride = gridDim.x * (blockDim.x >> 7);
    float s = 0.f, s2 = 0.f;
    for (int m = row; m < NENT; m += stride) {
        float v = AGG[(size_t)m * DHID + c];
        s += v; s2 += v * v;
    }
    atomicAdd(&stats[c], s);
    atomicAdd(&stats[128 + c], s2);
}

// x_out = tanh((agg - mean) * rsqrt(var+eps) * gamma + beta)
__global__ void bn_apply_kernel(const float* __restrict__ AGG,
                                const float* __restrict__ stats,
                                const float* __restrict__ gamma,
                                const float* __restrict__ beta,
                                float* __restrict__ Xout) {
    size_t i = (size_t)blockIdx.x * blockDim.x + threadIdx.x;
    if (i >= (size_t)NENT * DHID) return;
    int c = (int)(i & 127);
    const float inv_n = 1.0f / (float)NENT;
    float mean = stats[c] * inv_n;
    float var  = stats[128 + c] * inv_n - mean * mean;
    float v = (AGG[i] - mean) * rsqrtf(var + BN_EPS) * gamma[c] + beta[c];
    Xout[i] = tanhf(v);
}

// scores[t] = sum_d |x[h] + r[rel] - x[t]| ; one wave per triple
__global__ void score_kernel(const float* __restrict__ X,
                             const float* __restrict__ R,
                             const int* __restrict__ trip,
                             float* __restrict__ out) {
    int t = blockIdx.x * (blockDim.x >> 5) + (threadIdx.x >> 5);
    if (t >= NTRI) return;
    int lane = threadIdx.x & 31;
    int h  = trip[t * 3 + 0];
    int rl = trip[t * 3 + 1];
    int tl = trip[t * 3 + 2];
    const float4 hv = *(const float4*)(X + (size_t)h  * DHID + lane * 4);
    const float4 rv = *(const float4*)(R + (size_t)rl * DHID + lane * 4);
    const float4 tv = *(const float4*)(X + (size_t)tl * DHID + lane * 4);
    float s = fabsf(hv.x + rv.x - tv.x) + fabsf(hv.y + rv.y - tv.y) +
              fabsf(hv.z + rv.z - tv.z) + fabsf(hv.w + rv.w - tv.w);
    for (int off = 16; off > 0; off >>= 1) s += __shfl_down(s, off, 32);
    if (lane == 0) out[t] = s;
}

// ---------------------------------------------------------------------------
extern "C" void kernel_launch(void* const* d_in, const int* in_sizes, int n_in,
                              void* d_out, int out_size, void* d_ws, size_t ws_size,
                              hipStream_t stream) {
    (void)in_sizes; (void)n_in; (void)out_size; (void)ws_size;
    const float* ent   = (const float*)d_in[0];
    const float* bases = (const float*)d_in[1];
    const float* coeff = (const float*)d_in[2];
    const float* selfr = (const float*)d_in[3];
    const float* W1    = (const float*)d_in[4];
    const float* relw1 = (const float*)d_in[5];
    const float* g1    = (const float*)d_in[6];
    const float* b1    = (const float*)d_in[7];
    const float* W2    = (const float*)d_in[8];
    const float* relw2 = (const float*)d_in[9];
    const float* g2    = (const float*)d_in[10];
    const float* b2    = (const float*)d_in[11];
    // d_in[12] = ent_ids (identity gather, unused)
    const int* eidx  = (const int*)d_in[13];
    const int* etype = (const int*)d_in[14];
    const int* ey    = (const int*)d_in[15];
    const int* trip  = (const int*)d_in[16];
    float* out = (float*)d_out;

    float* ws    = (float*)d_ws;
    float* A     = ws + OFF_A;
    float* STATS = ws + OFF_STATS;
    float* AGG   = ws + OFF_AGG;
    float* X1    = ws + OFF_X1;
    float* X2    = ws + OFF_X2;
    float* R0    = ws + OFF_R0;
    float* R1    = ws + OFF_R1;
    float* R2    = ws + OFF_R2;

    const int relBlocks = (NRELV * DHID + 255) / 256;

    rel_init_kernel<<<relBlocks, 256, 0, stream>>>(coeff, bases, selfr, R0);

    // -------- layer 1
    zero_kernel<<<2048, 256, 0, stream>>>((float4*)A, (SZ_A + SZ_STATS) / 4);
    scatter_kernel<<<NEDGE / 8, 256, 0, stream>>>(ent, R0, eidx, etype, ey, A);
    gemm_wmma_kernel<<<(MTILES + 3) / 4, 128, GEMM_LDS_BYTES, stream>>>(A, W1, AGG);
    bn_stats_kernel<<<512, 256, 0, stream>>>(AGG, STATS);
    bn_apply_kernel<<<(NENT * DHID) / 256, 256, 0, stream>>>(AGG, STATS, g1, b1, X1);
    rel_mm_kernel<<<relBlocks, 256, 0, stream>>>(R0, relw1, R1);

    // -------- layer 2
    zero_kernel<<<2048, 256, 0, stream>>>((float4*)A, (SZ_A + SZ_STATS) / 4);
    scatter_kernel<<<NEDGE / 8, 256, 0, stream>>>(X1, R1, eidx, etype, ey, A);
    gemm_wmma_kernel<<<(MTILES + 3) / 4, 128, GEMM_LDS_BYTES, stream>>>(A, W2, AGG);
    bn_stats_kernel<<<512, 256, 0, stream>>>(AGG, STATS);
    bn_apply_kernel<<<(NENT * DHID) / 256, 256, 0, stream>>>(AGG, STATS, g2, b2, X2);
    rel_mm_kernel<<<relBlocks, 256, 0, stream>>>(R1, relw2, R2);

    // -------- scoring
    score_kernel<<<NTRI / 8, 256, 0, stream>>>(X2, R2, trip, out);
}